// MCUDetectionLoss_11768210391478
// MI455X (gfx1250) — compile-verified
//
#include <hip/hip_runtime.h>
#include <math.h>

// ---------------- problem constants (from setup_inputs) ----------------
#define NCLS   63
#define CCH    64          // 1 + NCLS channels in cls_p
#define BATCH  16
#define TT     32          // targets per image
#define H3     128
#define W3     128
#define H4     64
#define W4     64
#define CHUNK  4096        // floats handled per block
#define HALF_CHUNK 2048    // per TDM tile (8 KB), double-buffered

// ws float layout (every slot rewritten each launch -> replay-safe):
//  [0..4]   scale3 target partials: lb, lo_pos, lc, marked_sum, marked_cnt
//  [8..12]  scale4 target partials
//  [16..95] bg per-block partial sums (64 blocks scale3, 16 blocks scale4)
//  [96]     (as u32) ticket counter, zeroed by the targets kernel each launch

// ---------------- TDM availability probing (compile-time) ----------------
#if defined(__HIP_DEVICE_COMPILE__)
# if defined(__has_builtin)
#  if __has_builtin(__builtin_amdgcn_tensor_load_to_lds)
#   define USE_TDM 1
#  endif
# endif
#endif
#ifndef USE_TDM
# define USE_TDM 0
#endif

#if defined(__has_include)
# if __has_include(<hip/amd_detail/amd_gfx1250_TDM.h>)
#  define TDM_6ARG 1
# endif
#endif
#ifndef TDM_6ARG
# define TDM_6ARG 0
#endif

typedef unsigned int tdm_u32x4 __attribute__((ext_vector_type(4)));
typedef int          tdm_i32x4 __attribute__((ext_vector_type(4)));
typedef int          tdm_i32x8 __attribute__((ext_vector_type(8)));

// ---------------- math helpers (match jnp numerics) ----------------
__device__ __forceinline__ float sl1f(float d)  { d = fabsf(d); return d < 1.f ? 0.5f * d * d : d - 0.5f; }
__device__ __forceinline__ float bce1f(float x) { return fmaxf(x, 0.f) - x + log1pf(expf(-fabsf(x))); }   // y=1
__device__ __forceinline__ float bce0f(float x) { return fmaxf(x, 0.f)     + log1pf(expf(-fabsf(x))); }   // y=0
__device__ __forceinline__ float sigf(float x)  { return 1.f / (1.f + expf(-x)); }

__device__ __forceinline__ float block_reduce(float v, float* red) {
  const int tid = (int)threadIdx.x;
  red[tid] = v;
  __syncthreads();
  for (int s = (int)(blockDim.x >> 1); s > 0; s >>= 1) {
    if (tid < s) red[tid] += red[tid + s];
    __syncthreads();
  }
  float r = red[0];
  __syncthreads();
  return r;
}

#if USE_TDM
// Issue one TDM DMA: nelem contiguous f32 from `src` (global) into `lds_dst`.
// D# per CDNA5 ISA §8.3/8.4: count=1; data_size=4B; 2-D tile (nelem x 1);
// tensor dims == tile dims; workgroup_mask=0 (not in a cluster); type=2.
__device__ __forceinline__ void tdm_load_f32(const float* src, float* lds_dst, int nelem) {
  const unsigned long long ga = (unsigned long long)(uintptr_t)src;
  const unsigned int lds_off  = (unsigned int)(uintptr_t)lds_dst;  // low 32b of generic addr = LDS offset
  tdm_u32x4 g0 = { 1u, lds_off, (unsigned int)ga,
                   (unsigned int)(ga >> 32) | (2u << 30) };
  tdm_i32x8 g1 = { 0x20000,            // data_size=2 (4 bytes) << 16, mask=0
                   nelem << 16,        // tensor_dim0[15:0] in bits[31:16]
                   0x00010000,         // tensor_dim0 hi=0 | tensor_dim1=1
                   nelem << 16,        // tile_dim0 in bits[31:16]
                   1,                  // tile_dim1=1, tile_dim2=0
                   nelem,              // tensor_dim0_stride[31:0]
                   0, 0 };
  tdm_i32x4 gz = { 0, 0, 0, 0 };
#if TDM_6ARG
  tdm_i32x8 gz8 = { 0, 0, 0, 0, 0, 0, 0, 0 };
  __builtin_amdgcn_tensor_load_to_lds(g0, g1, gz, gz, gz8, 0);
#else
  __builtin_amdgcn_tensor_load_to_lds(g0, g1, gz, gz, 0);
#endif
}
#endif

// ---------------- kernel 1: per-target losses (both scales) + counter init ----------------
// grid = 2 blocks (block 0 -> scale3, block 1 -> scale4), 512 threads = B*T.
__global__ void det_targets_kernel(const float* __restrict__ cls3, const float* __restrict__ reg3,
                                   const float* __restrict__ t3,
                                   const float* __restrict__ cls4, const float* __restrict__ reg4,
                                   const float* __restrict__ t4,
                                   float* __restrict__ ws) {
  __shared__ float red[512];
  const int scale = (int)blockIdx.x;
  if (scale == 0 && threadIdx.x == 0) ((unsigned int*)ws)[96] = 0u;  // reset ticket for kernel 2

  const float* __restrict__ cls = scale ? cls4 : cls3;
  const float* __restrict__ reg = scale ? reg4 : reg3;
  const float* __restrict__ t   = scale ? t4   : t3;
  const int H = scale ? H4 : H3;
  const int W = scale ? W4 : W3;

  const int idx = (int)threadIdx.x;          // 0..511
  const int b = idx / TT, j = idx % TT;

  const float* tj = t + (size_t)(b * TT + j) * 5;
  const float cid_f = tj[0];
  const float tx = tj[1] * (float)W, ty = tj[2] * (float)H;
  const float tw = tj[3] * (float)W, th = tj[4] * (float)H;
  const int cls_id = (int)cid_f;
  const int gx = (int)fminf(fmaxf(tx, 0.f), (float)(W - 1));
  const int gy = (int)fminf(fmaxf(ty, 0.f), (float)(H - 1));

  // reg gather: reg_p[b, 0..3, gy, gx]
  const size_t hw = (size_t)H * W;
  const size_t rb = (size_t)b * 4 * hw + (size_t)gy * W + (size_t)gx;
  const float dx = sigf(reg[rb]);
  const float dy = sigf(reg[rb + hw]);
  const float dw = expf(fminf(fmaxf(reg[rb + 2 * hw], -4.f), 4.f));
  const float dh = expf(fminf(fmaxf(reg[rb + 3 * hw], -4.f), 4.f));
  const float px = (float)gx + dx, py = (float)gy + dy;

  float lb = sl1f((px - 0.5f * dw) - (tx - 0.5f * tw))
           + sl1f((py - 0.5f * dh) - (ty - 0.5f * th))
           + sl1f((px + 0.5f * dw) - (tx + 0.5f * tw))
           + sl1f((py + 0.5f * dh) - (ty + 0.5f * th));
  lb *= 0.25f;

  // obj + focal class loss: cls_p[b, :, gy, gx]
  const size_t cbase = (size_t)b * CCH * hw + (size_t)gy * W + (size_t)gx;
  const float obj_logit = cls[cbase];
  const float lo = bce1f(obj_logit);

  float lc = 0.f;
  for (int c = 0; c < NCLS; ++c) {
    const float x = cls[cbase + (size_t)(c + 1) * hw];
    const float y = (c == cls_id) ? 1.f : 0.f;
    const float bce = fmaxf(x, 0.f) - x * y + log1pf(expf(-fabsf(x)));
    const float p  = sigf(x);
    const float pt = p * y + (1.f - p) * (1.f - y);
    const float om = 1.f - pt;
    lc += 0.25f * om * om * bce;               // ALPHA * (1-pt)^GAMMA * bce
  }
  lc *= (1.f / (float)NCLS);                   // mean over classes

  // uniqueness: obj_map cell counted once even if several targets hit it
  bool unique = true;
  for (int jp = 0; jp < j; ++jp) {
    const float* tp = t + (size_t)(b * TT + jp) * 5;
    const int gxp = (int)fminf(fmaxf(tp[1] * (float)W, 0.f), (float)(W - 1));
    const int gyp = (int)fminf(fmaxf(tp[2] * (float)H, 0.f), (float)(H - 1));
    if (gxp == gx && gyp == gy) { unique = false; break; }
  }
  const float msum = unique ? bce0f(obj_logit) : 0.f;
  const float mcnt = unique ? 1.f : 0.f;

  // deterministic block tree-reductions, one slot-set per scale
  const float lb_t = block_reduce(lb, red);
  const float lo_t = block_reduce(lo, red);
  const float lc_t = block_reduce(lc, red);
  const float ms_t = block_reduce(msum, red);
  const float mc_t = block_reduce(mcnt, red);
  if (threadIdx.x == 0) {
    float* p = ws + scale * 8;
    p[0] = lb_t; p[1] = lo_t; p[2] = lc_t; p[3] = ms_t; p[4] = mc_t;
  }
}

// ---------------- kernel 2: background BCE + fused finalize ----------------
// 80 blocks: blk 0..63 -> scale3 (4 chunks of 4096 per image), blk 64..79 -> scale4.
// Each block issues TWO TDM DMAs (double-buffered 8 KB tiles), overlaps the
// reduction of tile A with the DMA of tile B using graduated TENSORcnt waits.
// Last block (atomic ticket) sums all partials in fixed order and writes out.
__global__ void det_bg_finalize_kernel(const float* __restrict__ cls3,
                                       const float* __restrict__ cls4,
                                       float* __restrict__ ws,
                                       float* __restrict__ out) {
  __shared__ float tileA[HALF_CHUNK];
  __shared__ float tileB[HALF_CHUNK];
  __shared__ float red[256];

  const int blk = (int)blockIdx.x;
  const float* __restrict__ src;
  if (blk < 64) {
    const int b = blk >> 2, part = blk & 3;
    src = cls3 + (size_t)b * CCH * (H3 * W3) + (size_t)part * CHUNK;   // channel-0 plane slice
  } else {
    const int b = blk - 64;
    src = cls4 + (size_t)b * CCH * (H4 * W4);                          // whole 64x64 plane
  }

#if USE_TDM
  if (threadIdx.x < 32) {  // one wave issues both DMAs (EXEC ignored; wave-level ops)
    tdm_load_f32(src,              tileA, HALF_CHUNK);
    tdm_load_f32(src + HALF_CHUNK, tileB, HALF_CHUNK);
    __builtin_amdgcn_s_wait_tensorcnt(1);   // tile A complete; tile B still in flight
  }
  __syncthreads();
  float s = 0.f;
  for (int i = (int)threadIdx.x; i < HALF_CHUNK; i += (int)blockDim.x)
    s += bce0f(tileA[i]);                   // overlaps with tile B DMA
  if (threadIdx.x < 32) __builtin_amdgcn_s_wait_tensorcnt(0);
  __syncthreads();
  for (int i = (int)threadIdx.x; i < HALF_CHUNK; i += (int)blockDim.x)
    s += bce0f(tileB[i]);
#else
  for (int i = (int)threadIdx.x; i < HALF_CHUNK; i += (int)blockDim.x) {
    tileA[i] = src[i];
    tileB[i] = src[HALF_CHUNK + i];
  }
  __syncthreads();
  float s = 0.f;
  for (int i = (int)threadIdx.x; i < CHUNK; i += (int)blockDim.x)
    s += bce0f(i < HALF_CHUNK ? tileA[i] : tileB[i - HALF_CHUNK]);
#endif

  const float tot = block_reduce(s, red);

  if (threadIdx.x == 0) {
    ws[16 + blk] = tot;
    __threadfence();
    unsigned int* ctr = (unsigned int*)ws + 96;
    const unsigned int prev = atomicAdd(ctr, 1u);
    if (prev == 79u) {                       // last block: finalize (fixed-order sums)
      __threadfence();
      const float lb = ws[0] + ws[8];
      float       lo = ws[1] + ws[9];
      const float lc = ws[2] + ws[10];
      float tot3 = 0.f, tot4 = 0.f;
      for (int i = 0;  i < 64; ++i) tot3 += ws[16 + i];
      for (int i = 64; i < 80; ++i) tot4 += ws[16 + i];
      const float bg3 = (tot3 - ws[3])  / fmaxf((float)(BATCH * H3 * W3) - ws[4],  1.f);
      const float bg4 = (tot4 - ws[11]) / fmaxf((float)(BATCH * H4 * W4) - ws[12], 1.f);
      lo += 0.05f * (bg3 + bg4);
      const float inv_n = 1.f / (float)(2 * BATCH * TT);   // n = n3 + n4 = 1024
      out[0] = (2.0f * lb + 1.0f * lo + 0.5f * lc) * inv_n;
    }
  }
}

// ---------------- host entry ----------------
extern "C" void kernel_launch(void* const* d_in, const int* in_sizes, int n_in,
                              void* d_out, int out_size, void* d_ws, size_t ws_size,
                              hipStream_t stream) {
  (void)in_sizes; (void)n_in; (void)out_size; (void)ws_size;
  const float* cls3 = (const float*)d_in[0];
  const float* reg3 = (const float*)d_in[1];
  const float* cls4 = (const float*)d_in[2];
  const float* reg4 = (const float*)d_in[3];
  const float* t3   = (const float*)d_in[4];
  const float* t4   = (const float*)d_in[5];
  float* ws  = (float*)d_ws;
  float* out = (float*)d_out;

  det_targets_kernel<<<dim3(2), dim3(512), 0, stream>>>(cls3, reg3, t3, cls4, reg4, t4, ws);
  det_bg_finalize_kernel<<<dim3(80), dim3(256), 0, stream>>>(cls3, cls4, ws, out);
}